// DQN_84808424227050
// MI455X (gfx1250) — compile-verified
//
#include <hip/hip_runtime.h>
#include <hip/hip_bf16.h>
#include <cstdint>
#include <cstddef>

typedef float v2f __attribute__((ext_vector_type(2)));
typedef float v8f __attribute__((ext_vector_type(8)));

// ---------------- utility ----------------
__global__ void fill_f(float* __restrict__ p, float v, long n) {
  long i = (long)blockIdx.x * blockDim.x + threadIdx.x;
  long str = (long)gridDim.x * blockDim.x;
  for (; i < n; i += str) p[i] = v;
}

__global__ void deg_count(const int* __restrict__ dst, float* __restrict__ deg, int E) {
  int e = blockIdx.x * blockDim.x + threadIdx.x;
  if (e < E) atomicAdd(&deg[dst[e]], 1.0f);
}

__global__ void rsqrt_ip(float* __restrict__ p, int n) {
  int i = blockIdx.x * blockDim.x + threadIdx.x;
  if (i < n) p[i] = rsqrtf(p[i]);
}

// ---------------- per-node GEMM: y[N,16] = x[N,K] @ W[K,16] ----------------
template <int K>
__global__ void node_gemm(const float* __restrict__ x, const float* __restrict__ W,
                          float* __restrict__ y, int N) {
  __shared__ float w[K * 16];
  for (int i = threadIdx.x; i < K * 16; i += blockDim.x) w[i] = W[i];
  __syncthreads();
  int n = blockIdx.x * blockDim.x + threadIdx.x;
  if (n >= N) return;
  const float* xr = x + (size_t)n * K;
  float acc[16];
#pragma unroll
  for (int d = 0; d < 16; ++d) acc[d] = 0.f;
  for (int k = 0; k < K; ++k) {
    float f = xr[k];
#pragma unroll
    for (int d = 0; d < 16; ++d) acc[d] += f * w[k * 16 + d];
  }
  float* yr = y + (size_t)n * 16;
#pragma unroll
  for (int d = 0; d < 16; ++d) yr[d] = acc[d];
}

// out[n,d] = y[n,d]*dinv[n]^2 + bias[d]   (self-loop term + bias)
__global__ void self_init(const float* __restrict__ y, const float* __restrict__ dinv,
                          const float* __restrict__ bias, float* __restrict__ out, int N) {
  int t = blockIdx.x * blockDim.x + threadIdx.x;
  if (t >= N * 16) return;
  int n = t >> 4, d = t & 15;
  float di = dinv[n];
  out[t] = y[t] * di * di + bias[d];
}

// out[dst] += y[src] * dinv[src]*dinv[dst]
__global__ void edge_scatter(const int* __restrict__ src, const int* __restrict__ dst,
                             const float* __restrict__ dinv, const float* __restrict__ y,
                             float* __restrict__ out, int E) {
  int e = blockIdx.x * blockDim.x + threadIdx.x;
  if (e >= E) return;
  int s = src[e], t = dst[e];
  float c = dinv[s] * dinv[t];
  const float4* yr = (const float4*)(y + (size_t)s * 16);
  float* o = out + (size_t)t * 16;
#pragma unroll
  for (int q = 0; q < 4; ++q) {
    float4 v = yr[q];
    atomicAdd(o + 4 * q + 0, v.x * c);
    atomicAdd(o + 4 * q + 1, v.y * c);
    atomicAdd(o + 4 * q + 2, v.z * c);
    atomicAdd(o + 4 * q + 3, v.w * c);
  }
}

// ---------------- multi-hop frontier counts (B=4 packed as float4) ----------------
__global__ void set_targets(const int* __restrict__ tgt, float* __restrict__ f, int B) {
  int b = blockIdx.x * blockDim.x + threadIdx.x;
  if (b < B) f[(size_t)tgt[b] * 4 + b] = 1.0f;
}

__global__ void hop_edge(const int* __restrict__ src, const int* __restrict__ dst,
                         const float* __restrict__ f, float* __restrict__ g, int E) {
  int e = blockIdx.x * blockDim.x + threadIdx.x;
  if (e >= E) return;
  int s = src[e], t = dst[e];
  float4 fs = ((const float4*)f)[s];
  float4 fd = ((const float4*)f)[t];
  if (fs.x != 0.f) atomicAdd(&g[(size_t)t * 4 + 0], fs.x);
  if (fs.y != 0.f) atomicAdd(&g[(size_t)t * 4 + 1], fs.y);
  if (fs.z != 0.f) atomicAdd(&g[(size_t)t * 4 + 2], fs.z);
  if (fs.w != 0.f) atomicAdd(&g[(size_t)t * 4 + 3], fs.w);
  if (fd.x != 0.f) atomicAdd(&g[(size_t)s * 4 + 0], fd.x);
  if (fd.y != 0.f) atomicAdd(&g[(size_t)s * 4 + 1], fd.y);
  if (fd.z != 0.f) atomicAdd(&g[(size_t)s * 4 + 2], fd.z);
  if (fd.w != 0.f) atomicAdd(&g[(size_t)s * 4 + 3], fd.w);
}

// total += g; f = g; g = 0
__global__ void hop_accum(float* __restrict__ total, float* __restrict__ f,
                          float* __restrict__ g, int n) {
  int i = blockIdx.x * blockDim.x + threadIdx.x;
  if (i >= n) return;
  float v = g[i];
  total[i] += v;
  f[i] = v;
  g[i] = 0.f;
}

// pooled[b,d] += total[b,n]*mu[n,d]; denom[b] += total[b,n]   (skip zeros)
__global__ void pool_reduce(const float* __restrict__ total, const float* __restrict__ mu,
                            float* __restrict__ pooled, float* __restrict__ denom, int N) {
  int n = blockIdx.x * blockDim.x + threadIdx.x;
  if (n >= N) return;
  float4 t4 = ((const float4*)total)[n];
  if (t4.x == 0.f && t4.y == 0.f && t4.z == 0.f && t4.w == 0.f) return;
  float tb[4] = {t4.x, t4.y, t4.z, t4.w};
  float m[16];
#pragma unroll
  for (int d = 0; d < 16; ++d) m[d] = mu[(size_t)n * 16 + d];
#pragma unroll
  for (int b = 0; b < 4; ++b) {
    if (tb[b] != 0.f) {
      atomicAdd(&denom[b], tb[b]);
#pragma unroll
      for (int d = 0; d < 16; ++d) atomicAdd(&pooled[b * 16 + d], tb[b] * m[d]);
    }
  }
}

// base[b,h] = fc1_b[h] + (pooled[b]/denom) @ fc1_w[0:16] + mu[target_b] @ fc1_w[16:32]
__global__ void base_kernel(const float* __restrict__ pooled, const float* __restrict__ denom,
                            const float* __restrict__ mu, const int* __restrict__ tgt,
                            const float* __restrict__ fc1_w, const float* __restrict__ fc1_b,
                            float* __restrict__ base, int B) {
  int t = blockIdx.x * blockDim.x + threadIdx.x;
  if (t >= B * 128) return;
  int b = t >> 7, h = t & 127;
  float dn = denom[b];
  dn = dn > 0.f ? dn : 1.0f;
  float s = fc1_b[h];
#pragma unroll
  for (int j = 0; j < 16; ++j) s += (pooled[b * 16 + j] / dn) * fc1_w[j * 128 + h];
  int tn = tgt[b];
#pragma unroll
  for (int j = 0; j < 16; ++j) s += mu[(size_t)tn * 16 + j] * fc1_w[(16 + j) * 128 + h];
  base[t] = s;
}

// ---------------- Q-head MLP via V_WMMA_F32_16X16X4_F32 ----------------
// h = relu(base_b + mu_a[16-row tile] @ fc1_w[32:48,:]);  q = h @ fc2_w + fc2_b
// One wave per 16-action tile, grid-strided. K=16 via 4 chained K=4 WMMAs,
// H=128 via 8 N-tiles, accumulator seeded with base (column-replicated == C layout).
__global__ __launch_bounds__(256) void mlp_wmma(
    const float* __restrict__ mu, const int* __restrict__ actions,
    const float* __restrict__ fc1_w, const float* __restrict__ base,
    const float* __restrict__ fc2_w, const float* __restrict__ fc2_b,
    float* __restrict__ q, int A, int tilesPerB, int ntiles) {
  const int lane = threadIdx.x & 31;
  const int hi = lane >> 4;  // half-wave: selects K pair {2hi, 2hi+1} within each K4 group
  const int lo = lane & 15;  // row (A) / column (B,C,D) index
  const int wave = (int)((blockIdx.x * blockDim.x + threadIdx.x) >> 5);
  const int nwaves = (int)((gridDim.x * blockDim.x) >> 5);

  // B fragments: fc1_w rows 32..47 (16 x 128), loaded once per wave.
  // Layout mirrors the documented A(16x4) layout: VGPR0 holds K={4j+2hi}, VGPR1 K={4j+2hi+1}.
  v2f Breg[8][4];
#pragma unroll
  for (int n = 0; n < 8; ++n)
#pragma unroll
    for (int j = 0; j < 4; ++j) {
      int row = 32 + 4 * j + 2 * hi;
      int col = 16 * n + lo;
      Breg[n][j].x = fc1_w[row * 128 + col];
      Breg[n][j].y = fc1_w[(row + 1) * 128 + col];
    }
  const float qb = fc2_b[0];

  for (int t = wave; t < ntiles; t += nwaves) {
    int b = t / tilesPerB;
    int a0 = (t - b * tilesPerB) * 16;

    int act = actions[(size_t)b * A + a0 + lo];  // row M = lo (both halves)
    float sgn = act < 0 ? -1.f : 1.f;
    int idx = act < 0 ? -act : act;
    const float* mrow = mu + (size_t)idx * 16;

    v2f Areg[4];
#pragma unroll
    for (int j = 0; j < 4; ++j) {
      v2f v = *(const v2f*)(mrow + 4 * j + 2 * hi);
      Areg[j].x = v.x * sgn;
      Areg[j].y = v.y * sgn;
    }

    float qacc[8];
#pragma unroll
    for (int i = 0; i < 8; ++i) qacc[i] = 0.f;

#pragma unroll
    for (int n = 0; n < 8; ++n) {
      float bval = base[b * 128 + 16 * n + lo];  // same for every row of this column
      v8f c;
#pragma unroll
      for (int i = 0; i < 8; ++i) c[i] = bval;
#pragma unroll
      for (int j = 0; j < 4; ++j)
        c = __builtin_amdgcn_wmma_f32_16x16x4_f32(false, Areg[j], false, Breg[n][j],
                                                  (short)0, c, false, false);
      float w2 = fc2_w[16 * n + lo];
#pragma unroll
      for (int i = 0; i < 8; ++i) {
        float h = c[i];
        h = h > 0.f ? h : 0.f;  // relu
        qacc[i] += h * w2;      // partial fc2 dot (this lane's column)
      }
    }

    // reduce each row's partial across its 16-lane half; lane lo==0 writes.
#pragma unroll
    for (int i = 0; i < 8; ++i) {
      float r = qacc[i];
      r += __shfl_xor(r, 1, 32);
      r += __shfl_xor(r, 2, 32);
      r += __shfl_xor(r, 4, 32);
      r += __shfl_xor(r, 8, 32);
      if (lo == 0) q[(size_t)b * A + a0 + i + 8 * hi] = r + qb;
    }
  }
}

// ---------------- driver ----------------
static inline int cdiv(long a, int b) { return (int)((a + b - 1) / b); }

extern "C" void kernel_launch(void* const* d_in, const int* in_sizes, int n_in,
                              void* d_out, int out_size, void* d_ws, size_t ws_size,
                              hipStream_t stream) {
  const float* feat  = (const float*)d_in[0];
  const float* W0    = (const float*)d_in[1];
  const float* b0    = (const float*)d_in[2];
  const float* W1    = (const float*)d_in[3];
  const float* b1    = (const float*)d_in[4];
  const float* fc1_w = (const float*)d_in[5];
  const float* fc1_b = (const float*)d_in[6];
  const float* fc2_w = (const float*)d_in[7];
  const float* fc2_b = (const float*)d_in[8];
  const int* eidx    = (const int*)d_in[9];
  const int* tgt     = (const int*)d_in[10];
  const int* acts    = (const int*)d_in[11];

  const int F = 64;
  const int N = in_sizes[0] / F;
  const int E = in_sizes[9] / 2;
  const int B = in_sizes[10];
  const int A = in_sizes[11] / B;
  const int* src = eidx;
  const int* dst = eidx + E;

  float* ws = (float*)d_ws;
  float* dinv   = ws; ws += N;
  float* y      = ws; ws += (size_t)N * 16;
  float* x1     = ws; ws += (size_t)N * 16;
  float* mu     = ws; ws += (size_t)N * 16;
  float* f      = ws; ws += (size_t)N * 4;
  float* g      = ws; ws += (size_t)N * 4;
  float* tot    = ws; ws += (size_t)N * 4;
  float* pooled = ws; ws += (size_t)B * 16;  // denom directly follows
  float* denom  = ws; ws += (size_t)B;
  float* base   = ws; ws += (size_t)B * 128;

  const int T = 256;

  // symmetric-norm degrees (dst-count + self loop), then deg^-1/2 in place
  fill_f<<<cdiv(N, T), T, 0, stream>>>(dinv, 1.0f, N);
  deg_count<<<cdiv(E, T), T, 0, stream>>>(dst, dinv, E);
  rsqrt_ip<<<cdiv(N, T), T, 0, stream>>>(dinv, N);

  // GCN layer 0: x1 = gcn(feat @ W0)
  node_gemm<64><<<cdiv(N, T), T, 0, stream>>>(feat, W0, y, N);
  self_init<<<cdiv((long)N * 16, T), T, 0, stream>>>(y, dinv, b0, x1, N);
  edge_scatter<<<cdiv(E, T), T, 0, stream>>>(src, dst, dinv, y, x1, E);

  // GCN layer 1: mu = gcn(x1 @ W1)
  node_gemm<16><<<cdiv(N, T), T, 0, stream>>>(x1, W1, y, N);
  self_init<<<cdiv((long)N * 16, T), T, 0, stream>>>(y, dinv, b1, mu, N);
  edge_scatter<<<cdiv(E, T), T, 0, stream>>>(src, dst, dinv, y, mu, E);

  // 2-hop multiplicity counts for B targets
  fill_f<<<cdiv((long)N * 4, T), T, 0, stream>>>(f, 0.f, (long)N * 4);
  fill_f<<<cdiv((long)N * 4, T), T, 0, stream>>>(g, 0.f, (long)N * 4);
  fill_f<<<cdiv((long)N * 4, T), T, 0, stream>>>(tot, 0.f, (long)N * 4);
  set_targets<<<1, 32, 0, stream>>>(tgt, f, B);
  for (int h = 0; h < 2; ++h) {
    hop_edge<<<cdiv(E, T), T, 0, stream>>>(src, dst, f, g, E);
    hop_accum<<<cdiv((long)N * 4, T), T, 0, stream>>>(tot, f, g, N * 4);
  }

  // pooled embedding + per-batch base vector
  fill_f<<<1, T, 0, stream>>>(pooled, 0.f, (long)B * 16 + B);
  pool_reduce<<<cdiv(N, T), T, 0, stream>>>(tot, mu, pooled, denom, N);
  base_kernel<<<cdiv((long)B * 128, T), T, 0, stream>>>(pooled, denom, mu, tgt, fc1_w, fc1_b,
                                                        base, B);

  // WMMA Q-head over all B*A actions
  int tilesPerB = A / 16;
  int ntiles = B * tilesPerB;
  mlp_wmma<<<512, 256, 0, stream>>>(mu, acts, fc1_w, base, fc2_w, fc2_b, (float*)d_out, A,
                                    tilesPerB, ntiles);
}